// EdgeSAGE_566935683375
// MI455X (gfx1250) — compile-verified
//
#include <hip/hip_runtime.h>
#include <hip/hip_bf16.h>
#include <math.h>

typedef __attribute__((ext_vector_type(2))) float v2f;
typedef __attribute__((ext_vector_type(8))) float v8f;

#define D 128

// ---------------------------------------------------------------------------
// Zero a float buffer (grid-stride).
// ---------------------------------------------------------------------------
__global__ __launch_bounds__(256) void zero_kernel(float* p, long long n) {
    long long i = (long long)blockIdx.x * blockDim.x + threadIdx.x;
    long long stride = (long long)gridDim.x * blockDim.x;
    for (; i < n; i += stride) p[i] = 0.0f;
}

// ---------------------------------------------------------------------------
// Edge aggregation: s[dst] += x[src], cnt[dst] += 1.  One wave32 per edge,
// each lane moves a float4 (wave reads a full coalesced 512B row) and issues
// 4 global_atomic_add_f32 that resolve in L2 (s fits in the 192MB L2).
// ---------------------------------------------------------------------------
__global__ __launch_bounds__(256) void aggregate_kernel(
    const float* __restrict__ x, const int* __restrict__ src,
    const int* __restrict__ dst, float* __restrict__ s,
    float* __restrict__ cnt, int E)
{
    int wid  = (int)(((long long)blockIdx.x * blockDim.x + threadIdx.x) >> 5);
    int lane = threadIdx.x & 31;
    if (wid >= E) return;                      // uniform per-wave exit
    int si = src[wid];
    int di = dst[wid];
    float4 v = ((const float4*)(x + (long long)si * D))[lane];
    float* so = s + (long long)di * D + lane * 4;
    atomicAdd(so + 0, v.x);
    atomicAdd(so + 1, v.y);
    atomicAdd(so + 2, v.z);
    atomicAdd(so + 3, v.w);
    if (lane == 0) atomicAdd(cnt + di, 1.0f);
}

// ---------------------------------------------------------------------------
// Fused SAGE linear: out = (s/max(cnt,1)) @ Wl + b + x @ Wr, optional ReLU.
// fp32 WMMA (V_WMMA_F32_16X16X4_F32). One wave computes a 16x16 output tile;
// block = 8 waves covers one group of 16 node-rows x all 128 output columns.
//
// 32-bit A 16x4 fragment: lane L<16 -> row M=L holds K=k+0,k+1;
//                         lane L>=16 -> row M=L-16 holds K=k+2,k+3.
// 32-bit B 4x16 fragment: v0 = B[k + (L<16?0:2)][n], v1 = B[k + (L<16?1:3)][n].
// C 16x16: value r on lane L is element (M = r + 8*(L>>4), N = L&15).
// ---------------------------------------------------------------------------
__global__ __launch_bounds__(256) void sage_gemm_kernel(
    const float* __restrict__ xin, const float* __restrict__ s,
    const float* __restrict__ cnt, const float* __restrict__ Wl,
    const float* __restrict__ bias, const float* __restrict__ Wr,
    float* __restrict__ out, int N, int relu)
{
    int mtile = blockIdx.x;                    // node-row tile (16 rows)
    int wave  = threadIdx.x >> 5;              // 0..7 -> output-column tile
    int lane  = threadIdx.x & 31;
    int m0    = mtile * 16;
    int n0    = wave * 16;
    int ncol  = n0 + (lane & 15);

    int rowA  = m0 + (lane & 15);
    if (rowA > N - 1) rowA = N - 1;            // clamp loads; keep EXEC all-1s
    int khalf = (lane >> 4) << 1;              // 0 or 2

    const float* srow = s   + (long long)rowA * D;
    const float* xrow = xin + (long long)rowA * D;
    float inv = 1.0f / fmaxf(cnt[rowA], 1.0f);

    v8f c = {};
    #pragma unroll
    for (int k = 0; k < D; k += 4) {
        int ka = k + khalf;
        // aggr @ Wl
        v2f a0; a0.x = srow[ka] * inv;  a0.y = srow[ka + 1] * inv;
        v2f b0; b0.x = Wl[ka * D + ncol]; b0.y = Wl[(ka + 1) * D + ncol];
        c = __builtin_amdgcn_wmma_f32_16x16x4_f32(
                false, a0, false, b0, (short)0, c, false, false);
        // x @ Wr
        v2f a1; a1.x = xrow[ka];        a1.y = xrow[ka + 1];
        v2f b1; b1.x = Wr[ka * D + ncol]; b1.y = Wr[(ka + 1) * D + ncol];
        c = __builtin_amdgcn_wmma_f32_16x16x4_f32(
                false, a1, false, b1, (short)0, c, false, false);
    }

    float bn = bias[ncol];
    int mbase = m0 + ((lane >> 4) << 3);       // +0 or +8
    #pragma unroll
    for (int r = 0; r < 8; ++r) {
        int m = mbase + r;
        if (m < N) {
            float v = c[r] + bn;
            if (relu) v = fmaxf(v, 0.0f);
            out[(long long)m * D + ncol] = v;
        }
    }
}

// ---------------------------------------------------------------------------
// Decode: cosine similarity + sigmoid.  One wave32 per pair; float4 loads,
// shfl_xor tree reduction across the wave.
// ---------------------------------------------------------------------------
__global__ __launch_bounds__(256) void decode_kernel(
    const float* __restrict__ h, const int* __restrict__ dec,
    float* __restrict__ out, int Dk)
{
    int wid  = (int)(((long long)blockIdx.x * blockDim.x + threadIdx.x) >> 5);
    int lane = threadIdx.x & 31;
    if (wid >= Dk) return;                     // uniform per-wave exit
    int ai = dec[wid];
    int bi = dec[Dk + wid];
    float4 a = ((const float4*)(h + (long long)ai * D))[lane];
    float4 b = ((const float4*)(h + (long long)bi * D))[lane];
    float dot = a.x * b.x + a.y * b.y + a.z * b.z + a.w * b.w;
    float na  = a.x * a.x + a.y * a.y + a.z * a.z + a.w * a.w;
    float nb  = b.x * b.x + b.y * b.y + b.z * b.z + b.w * b.w;
    #pragma unroll
    for (int off = 16; off > 0; off >>= 1) {
        dot += __shfl_xor(dot, off, 32);
        na  += __shfl_xor(na,  off, 32);
        nb  += __shfl_xor(nb,  off, 32);
    }
    if (lane == 0) {
        float denom = fmaxf(sqrtf(na) * sqrtf(nb), 1e-6f);
        float z = dot / denom;
        out[wid] = 1.0f / (1.0f + expf(-z));
    }
}

// ---------------------------------------------------------------------------
extern "C" void kernel_launch(void* const* d_in, const int* in_sizes, int n_in,
                              void* d_out, int out_size, void* d_ws, size_t ws_size,
                              hipStream_t stream) {
    const float* x   = (const float*)d_in[0];
    const int*   edg = (const int*)d_in[1];
    const int*   dec = (const int*)d_in[2];
    const float* W1l = (const float*)d_in[3];
    const float* b1  = (const float*)d_in[4];
    const float* W1r = (const float*)d_in[5];
    const float* W2l = (const float*)d_in[6];
    const float* b2  = (const float*)d_in[7];
    const float* W2r = (const float*)d_in[8];
    float* out = (float*)d_out;

    const int N  = in_sizes[0] / D;    // 50000
    const int E  = in_sizes[1] / 2;    // 1.6M
    const int Dk = in_sizes[2] / 2;    // 500k
    const int* src = edg;
    const int* dst = edg + E;

    float* ws  = (float*)d_ws;
    float* s   = ws;                           // N*D
    float* cnt = s + (long long)N * D;         // N
    float* h1  = cnt + N;                      // N*D
    float* h2  = h1 + (long long)N * D;        // N*D

    const long long sAndCnt = (long long)N * D + N;
    const int zeroBlocks  = 2048;
    const int aggrBlocks  = (E + 7) / 8;       // 1 wave per edge, 8 waves/block
    const int gemmBlocks  = (N + 15) / 16;
    const int decBlocks   = (Dk + 7) / 8;

    // ---- layer 1 ----
    zero_kernel<<<zeroBlocks, 256, 0, stream>>>(s, sAndCnt);
    aggregate_kernel<<<aggrBlocks, 256, 0, stream>>>(x, src, dst, s, cnt, E);
    sage_gemm_kernel<<<gemmBlocks, 256, 0, stream>>>(x, s, cnt, W1l, b1, W1r,
                                                     h1, N, /*relu=*/1);
    // ---- layer 2 (reuse s/cnt) ----
    zero_kernel<<<zeroBlocks, 256, 0, stream>>>(s, sAndCnt);
    aggregate_kernel<<<aggrBlocks, 256, 0, stream>>>(h1, src, dst, s, cnt, E);
    sage_gemm_kernel<<<gemmBlocks, 256, 0, stream>>>(h1, s, cnt, W2l, b2, W2r,
                                                     h2, N, /*relu=*/0);
    // ---- decode ----
    decode_kernel<<<decBlocks, 256, 0, stream>>>(h2, dec, out, Dk);
}